// DynamicMemoryCell_11888469475688
// MI455X (gfx1250) — compile-verified
//
#include <hip/hip_runtime.h>
#include <hip/hip_bf16.h>
#include <math.h>

typedef __attribute__((ext_vector_type(2))) float v2f;
typedef __attribute__((ext_vector_type(4))) float v4f;
typedef __attribute__((ext_vector_type(8))) float v8f;

#define LDSTR 68   // padded row stride (floats) for U/V in LDS: bank-conflict-free b64 reads

__global__ __launch_bounds__(256) void dmc_kernel(
    const float* __restrict__ s,
    const float* __restrict__ h,
    const float* __restrict__ keys,
    const float* __restrict__ U,
    const float* __restrict__ V,
    const float* __restrict__ W,
    const float* __restrict__ bias,
    const float* __restrict__ prelu_a,
    float* __restrict__ out,
    int n_tiles)
{
    __shared__ float Ul[64 * LDSTR];
    __shared__ float Vl[64 * LDSTR];
    __shared__ float sl[64];
    __shared__ float tl[64];

    const int tid = threadIdx.x;

    // ---- stage s into LDS ----
    if (tid < 64) sl[tid] = s[tid];

    // ---- stage U, V into LDS (padded rows), float4 global loads ----
    #pragma unroll
    for (int it = 0; it < 4; ++it) {
        int idx4 = tid + it * 256;          // 1024 float4 per matrix
        int flat = idx4 * 4;
        int row  = flat >> 6;
        int col  = flat & 63;
        v4f u = *(const v4f*)(U + flat);
        v4f v = *(const v4f*)(V + flat);
        *(v4f*)(&Ul[row * LDSTR + col]) = u;
        *(v4f*)(&Vl[row * LDSTR + col]) = v;
    }

    // ---- t = W @ s + bias (row-invariant), once per block ----
    if (tid < 64) {
        float acc = bias[tid];
        const float* wr = W + tid * 64;
        #pragma unroll 8
        for (int k = 0; k < 64; ++k) acc += wr[k] * s[k];
        tl[tid] = acc;
    }
    __syncthreads();

    const int wave = tid >> 5;
    const int lane = tid & 31;
    const int mrow = lane & 15;
    const int half = lane >> 4;

    const int tile = blockIdx.x * 8 + wave;   // wave-uniform -> EXEC stays all-ones
    if (tile >= n_tiles) return;

    const size_t rowbase = (size_t)tile * 16;

    // ---- load A-fragments for hb and keys (16x4 f32 layout) ----
    const float* hrow = h    + (rowbase + mrow) * 64 + half * 2;
    const float* krow = keys + (rowbase + mrow) * 64 + half * 2;
    v2f ah[16], ak[16];
    #pragma unroll
    for (int kk = 0; kk < 16; ++kk) {
        ah[kk] = *(const v2f*)(hrow + kk * 4);
        ak[kk] = *(const v2f*)(krow + kk * 4);
    }

    // ---- gate: dot(hb+keys, s) per row, halves hold disjoint K subsets ----
    float dp = 0.f;
    #pragma unroll
    for (int kk = 0; kk < 16; ++kk) {
        v2f sv = *(const v2f*)(&sl[kk * 4 + half * 2]);
        dp += (ah[kk].x + ak[kk].x) * sv.x + (ah[kk].y + ak[kk].y) * sv.y;
    }
    dp += __shfl_xor(dp, 16, 32);
    const float g = 1.0f / (1.0f + __expf(-dp));

    // ---- pre = hb@U^T + keys@V^T via V_WMMA_F32_16X16X4_F32 ----
    v8f C[4];
    #pragma unroll
    for (int n = 0; n < 4; ++n) {
        v8f c = {};
        const int rb = (n * 16 + mrow) * LDSTR + half * 2;
        #pragma unroll
        for (int kk = 0; kk < 16; ++kk) {
            v2f bu = *(const v2f*)(&Ul[rb + kk * 4]);
            v2f bv = *(const v2f*)(&Vl[rb + kk * 4]);
            c = __builtin_amdgcn_wmma_f32_16x16x4_f32(false, ah[kk], false, bu,
                                                      (short)0, c, false, false);
            c = __builtin_amdgcn_wmma_f32_16x16x4_f32(false, ak[kk], false, bv,
                                                      (short)0, c, false, false);
        }
        C[n] = c;
    }

    // ---- epilogue: PReLU, gated residual, row L2-normalize, store ----
    const float pa = prelu_a[0];
    float tc[4];
    #pragma unroll
    for (int n = 0; n < 4; ++n) tc[n] = tl[n * 16 + mrow];

    #pragma unroll
    for (int j = 0; j < 8; ++j) {
        const int m = j + 8 * half;                     // this lane's output row
        const float gm = __shfl(g, m, 16);              // g lives at lane (m) in each half
        const float* hr = h + (rowbase + m) * 64 + mrow;
        float hn[4];
        float sq = 0.f;
        #pragma unroll
        for (int n = 0; n < 4; ++n) {
            float pre = C[n][j] + tc[n];
            float act = (pre >= 0.f) ? pre : pa * pre;
            float x = hr[n * 16] + gm * act;
            hn[n] = x;
            sq += x * x;
        }
        sq += __shfl_xor(sq, 1, 16);
        sq += __shfl_xor(sq, 2, 16);
        sq += __shfl_xor(sq, 4, 16);
        sq += __shfl_xor(sq, 8, 16);
        const float rn = 1.0f / sqrtf(sq);
        float* orow = out + (rowbase + m) * 64 + mrow;
        #pragma unroll
        for (int n = 0; n < 4; ++n) orow[n * 16] = hn[n] * rn;
    }
}

extern "C" void kernel_launch(void* const* d_in, const int* in_sizes, int n_in,
                              void* d_out, int out_size, void* d_ws, size_t ws_size,
                              hipStream_t stream) {
    const float* s       = (const float*)d_in[0];
    const float* h       = (const float*)d_in[1];
    const float* keys    = (const float*)d_in[2];
    const float* U       = (const float*)d_in[3];
    const float* V       = (const float*)d_in[4];
    const float* W       = (const float*)d_in[5];
    const float* bias    = (const float*)d_in[6];
    const float* prelu_a = (const float*)d_in[7];
    float* out = (float*)d_out;

    const int N = in_sizes[1] / 64;        // 500000 rows
    const int n_tiles = N / 16;            // 31250 full 16-row tiles
    const int blocks = (n_tiles + 7) / 8;  // 8 waves per 256-thread block

    hipLaunchKernelGGL(dmc_kernel, dim3(blocks), dim3(256), 0, stream,
                       s, h, keys, U, V, W, bias, prelu_a, out, n_tiles);
}